// SSL_40973988004013
// MI455X (gfx1250) — compile-verified
//
#include <hip/hip_runtime.h>
#include <hip/hip_bf16.h>
#include <stdint.h>

typedef __attribute__((ext_vector_type(2))) float v2f;
typedef __attribute__((ext_vector_type(8))) float v8f;

#define KDIM   4096
#define NDIM   4096
#define REDIN  2048
#define NCHUNK 128     // K / 32
#define BST    68      // lB row stride in dwords: 32 cols duplicated -> 64, +4 pad

// Async DMA: copy 16B from global to LDS, tracked by ASYNCcnt (gfx1250).
__device__ __forceinline__ void async_ld_b128(uint32_t lds_byte_off, const float* g) {
    asm volatile("global_load_async_to_lds_b128 %0, %1, off"
                 :: "v"(lds_byte_off), "v"(g)
                 : "memory");
}

__global__ __launch_bounds__(256) void ssl_wmma_gemm(
    const float* __restrict__ x, const float* __restrict__ w,
    const float* __restrict__ bias, const int* __restrict__ rn,
    float* __restrict__ out)
{
    // A: [buf][k/4][m][k&3] interleaved -> one conflict-free ds_load_b64 per frag
    __shared__ float lA[2][8 * 128 * 4];           // 2 x 16 KB
    // B: [buf][n][c] raw compressed tile, columns duplicated (c and c+32 identical)
    __shared__ float lB[2][128 * BST];             // 2 x 34 KB

    const int t     = threadIdx.x;
    const int lane  = t & 31;
    const int wave  = t >> 5;
    const int waveM = wave >> 2;          // 0..1 -> 64-row slab
    const int waveN = wave & 3;           // 0..3 -> 32-col slab
    const int m0    = blockIdx.y * 128;
    const int n0    = blockIdx.x * 128;

    const int R0 = rn[0], R1 = rn[1], R2 = rn[2], R3 = rn[3];

    const int tr = t >> 3;          // 0..31 : row group for cooperative loads
    const int tc = (t & 7) << 2;    // 0..28 step 4 : float4 column group
    const int p  = tc >> 2;

    const float* gA = x + (size_t)(m0 + tr) * KDIM  + tc;
    const float* gB = w + (size_t)(n0 + tr) * REDIN + tc;

    const uint32_t lAo[2] = { (uint32_t)(uintptr_t)&lA[0][0],
                              (uint32_t)(uintptr_t)&lA[1][0] };
    const uint32_t lBo[2] = { (uint32_t)(uintptr_t)&lB[0][0],
                              (uint32_t)(uintptr_t)&lB[1][0] };

    const uint32_t aCell = (uint32_t)(p * 512 + tr * 4) * 4u;   // + rep*512 bytes
    const uint32_t bCell = (uint32_t)(tr * BST + tc) * 4u;      // + rep*32*BST*4 bytes

    auto issueA = [&](int buf, int i) {
#pragma unroll
        for (int rep = 0; rep < 4; ++rep)
            async_ld_b128(lAo[buf] + aCell + rep * 512,
                          gA + (size_t)(rep * 32) * KDIM + (size_t)i * 32);
    };
    auto issueB = [&](int buf, int it) {
#pragma unroll
        for (int rep = 0; rep < 4; ++rep) {
            const float* g = gB + (size_t)(rep * 32) * REDIN + (size_t)it * 32;
            const uint32_t d = lBo[buf] + bCell + rep * 32 * BST * 4;
            async_ld_b128(d, g);               // columns [tc..tc+3]
            async_ld_b128(d + 32 * 4, g);      // duplicate at +32 (rotation wrap)
        }
    };

    v8f acc[4][2] = {};

    issueA(0, 0);
    issueB(0, 0);

    const int lm   = lane & 15;
    const int half = (lane >> 4) << 1;   // lanes 0-15 -> K+0/1, lanes 16-31 -> K+2/3

    for (int i = 0; i < NCHUNK; ++i) {
        const int it = i >> 1;

        if (i + 1 < NCHUNK) {
            issueA((i + 1) & 1, i + 1);
            if (((i + 1) & 1) == 0)                 // B changes only every 2 chunks
                issueB(((i + 1) >> 1) & 1, (i + 1) >> 1);
            // batch(i) done when remaining <= size(batch(i+1)); loads retire in order
            if (i & 1) asm volatile("s_wait_asynccnt 0xc" ::: "memory");
            else       asm volatile("s_wait_asynccnt 0x4" ::: "memory");
        } else {
            asm volatile("s_wait_asynccnt 0x0" ::: "memory");
        }
        __syncthreads();

        // SSL hash-rotation offset: wave-uniform for this chunk (scalar)
        const int nj   = (n0 >> 5) + waveN;
        const int hh   = R3 + R2 * nj + R1 * (it + 1) + R0 * ((i & 1) + 1);
        const int offB = 32 - ((hh * 4) & 31);      // 4..32, multiple of 4

        const float* __restrict__ pa = &lA[i & 1][0];
        const float* __restrict__ pb = &lB[it & 1][0];

#pragma unroll
        for (int kk = 0; kk < 32; kk += 4) {
            v2f af[4], bf[2];
#pragma unroll
            for (int mi = 0; mi < 4; ++mi) {
                const int m = waveM * 64 + mi * 16 + lm;
                af[mi] = *(const v2f*)&pa[(kk >> 2) * 512 + m * 4 + half];
            }
#pragma unroll
            for (int ni = 0; ni < 2; ++ni) {
                const int nrow = waveN * 32 + ni * 16 + lm;
                bf[ni] = *(const v2f*)&pb[nrow * BST + offB + kk + half];
            }
#pragma unroll
            for (int mi = 0; mi < 4; ++mi)
#pragma unroll
                for (int ni = 0; ni < 2; ++ni)
                    acc[mi][ni] = __builtin_amdgcn_wmma_f32_16x16x4_f32(
                        false, af[mi], false, bf[ni],
                        (short)0, acc[mi][ni], false, false);
        }
        __syncthreads();
    }

    // ---- epilogue: bias add + store (C layout: VGPR r = row, lane = col) ----
    const int rhi = (lane >> 4) * 8;
#pragma unroll
    for (int ni = 0; ni < 2; ++ni) {
        const int col = n0 + waveN * 32 + ni * 16 + lm;
        const float bv = bias[col];
#pragma unroll
        for (int mi = 0; mi < 4; ++mi)
#pragma unroll
            for (int r = 0; r < 8; ++r) {
                const int row = m0 + waveM * 64 + mi * 16 + rhi + r;
                out[(size_t)row * NDIM + col] = acc[mi][ni][r] + bv;
            }
    }
}

extern "C" void kernel_launch(void* const* d_in, const int* in_sizes, int n_in,
                              void* d_out, int out_size, void* d_ws, size_t ws_size,
                              hipStream_t stream) {
    const float* x    = (const float*)d_in[0];   // [4, 2048, 4096] f32
    const float* w    = (const float*)d_in[1];   // [4096, 2048] f32 (compressed)
    const float* bias = (const float*)d_in[2];   // [4096] f32
    const int*   rn   = (const int*)d_in[3];     // [4] int32
    float*       out  = (float*)d_out;           // [8192, 4096] f32

    const int M = in_sizes[0] / KDIM;            // 8192
    dim3 grid(NDIM / 128, M / 128);              // (32, 64)
    ssl_wmma_gemm<<<grid, 256, 0, stream>>>(x, w, bias, rn, out);
}